// CombinedMetricLoss_17875653886477
// MI455X (gfx1250) — compile-verified
//
#include <hip/hip_runtime.h>
#include <math.h>

typedef __attribute__((ext_vector_type(2))) float v2f;
typedef __attribute__((ext_vector_type(8))) float v8f;
typedef __attribute__((ext_vector_type(4))) unsigned int u32x4;
typedef __attribute__((ext_vector_type(8))) int i32x8;
typedef __attribute__((ext_vector_type(4))) int i32x4;

#define B_SZ 2048
#define D_SZ 512
#define C_SZ 50000
#define TM 128
#define TN 64
#define TK 32
#define LDSS 36               // 32 + 4 pad (TDM pad_interval=32 dw, pad_amount=4 dw); gcd(36,64)=4 -> conflict free
#define NCH (D_SZ / TK)       // 16 K-chunks
#define NT_ARC ((C_SZ + TN - 1) / TN)   // 782 column tiles
#define SCALE_S 30.0f
#define COS_M_C 0.87758256189037276f
#define SIN_M_C 0.47942553860420301f
#define TRIP_MARGIN 0.3f

#if __has_builtin(__builtin_amdgcn_tensor_load_to_lds) && __has_builtin(__builtin_amdgcn_s_wait_tensorcnt)
#define HAVE_TDM 1
#else
#define HAVE_TDM 0
#endif

#if HAVE_TDM
// ---------------------------------------------------------------------------
// Issue one TDM 2D tile load: global (row-major, row_stride elems) -> LDS,
// with 4-DWORD padding after each 32-DWORD row (LDS row stride = 36 floats).
// Rows >= tensor_d1 are zero-filled by TDM OOB handling.
// D# layout per CDNA5 ISA ch.8 (group0 128b, group1 256b; groups 2/3 zero).
// This toolchain exposes the 6-arg builtin: (g0, g1, g2, g3, g_extra, cpol).
// ---------------------------------------------------------------------------
__device__ __forceinline__ void tdm_load_2d(unsigned lds_off, const float* gptr,
                                            unsigned tensor_d1, unsigned tile_d1,
                                            unsigned row_stride_elems) {
    unsigned long long ga = (unsigned long long)(uintptr_t)gptr;
    u32x4 g0;
    g0[0] = 1u;                                        // count=1 (valid), user mode
    g0[1] = lds_off;                                   // lds_addr (bytes)
    g0[2] = (unsigned)(ga & 0xFFFFFFFFu);              // global_addr[31:0]
    g0[3] = (unsigned)((ga >> 32) & 0x01FFFFFFu)       // global_addr[56:32]
          | (2u << 30);                                // type=2 ("image")
    i32x8 g1;
    // data_size=2 (4B) | pad_enable | pad_interval=4 (32 dw) | pad_amount=3 (4 dw)
    g1[0] = (2 << 16) | (1 << 20) | (4 << 22) | (3 << 25);
    const unsigned td0 = TK;                           // tensor_dim0 = 32 (K fully valid)
    g1[1] = (int)((td0 & 0xFFFFu) << 16);              // [63:48] = tensor_dim0 lo16
    g1[2] = (int)((td0 >> 16) | ((tensor_d1 & 0xFFFFu) << 16)); // dim0 hi16 | dim1 lo16
    g1[3] = (int)((tensor_d1 >> 16) | ((unsigned)TK << 16));    // dim1 hi16 | tile_dim0=32
    g1[4] = (int)(tile_d1 & 0xFFFFu);                  // tile_dim1 | tile_dim2=0
    g1[5] = (int)row_stride_elems;                     // tensor_dim0_stride[31:0]
    g1[6] = 0;                                         // stride hi16 | dim1_stride lo16
    g1[7] = 0;
    i32x4 z4; z4[0] = 0; z4[1] = 0; z4[2] = 0; z4[3] = 0;
    i32x8 z8; z8[0] = 0; z8[1] = 0; z8[2] = 0; z8[3] = 0;
               z8[4] = 0; z8[5] = 0; z8[6] = 0; z8[7] = 0;
    __builtin_amdgcn_tensor_load_to_lds(g0, g1, z4, z4, z8, 0);
}
#endif

// ---------------------------------------------------------------------------
// Row norms: invn[row] = 1/||M[row]||, optionally sq[row] = ||M[row]||^2
// ---------------------------------------------------------------------------
__global__ __launch_bounds__(256) void rownorm_kernel(const float* __restrict__ M,
                                                      float* __restrict__ invn,
                                                      float* __restrict__ sq) {
    __shared__ float red[256];
    const int row = blockIdx.x;
    const int tid = threadIdx.x;
    float s = 0.f;
    for (int d = tid; d < D_SZ; d += 256) {
        float v = M[(size_t)row * D_SZ + d];
        s += v * v;
    }
    red[tid] = s;
    __syncthreads();
    for (int off = 128; off > 0; off >>= 1) {
        if (tid < off) red[tid] += red[tid + off];
        __syncthreads();
    }
    if (tid == 0) {
        float t = red[0];
        invn[row] = rsqrtf(fmaxf(t, 1e-30f));
        if (sq) sq[row] = t;
    }
}

// ---------------------------------------------------------------------------
// Shared GEMM inner machinery: 128(M) x 64(N) tile, 8 waves, fp32 WMMA,
// TDM double-buffered LDS staging (fallback: manual float4 staging).
// ---------------------------------------------------------------------------
#define GEMM_TILE_LOOP(A_PTR, B_PTR, B_TENSOR_D1)                                        \
    v8f acc[4] = {};                                                                     \
    {                                                                                    \
        _Pragma("unroll 1")                                                              \
        for (int ch = 0; ch < NCH; ++ch) {                                               \
            const int cur = ch & 1;                                                      \
            STAGE_WAIT(ch);                                                              \
            __syncthreads();                                                             \
            STAGE_ISSUE_NEXT(ch, A_PTR, B_PTR, B_TENSOR_D1);                             \
            const float* __restrict__ cA = &lA[cur][0];                                  \
            const float* __restrict__ cB = &lB[cur][0];                                  \
            const int mrow = wave * 16 + (lane & 15);                                    \
            const int koff = (lane >> 4) * 2;                                            \
            _Pragma("unroll 4")                                                          \
            for (int kk = 0; kk < TK; kk += 4) {                                         \
                v2f a = *(const v2f*)(&cA[mrow * LDSS + kk + koff]);                     \
                _Pragma("unroll")                                                        \
                for (int ni = 0; ni < 4; ++ni) {                                         \
                    v2f b = *(const v2f*)(&cB[(ni * 16 + (lane & 15)) * LDSS + kk + koff]); \
                    acc[ni] = __builtin_amdgcn_wmma_f32_16x16x4_f32(                     \
                        false, a, false, b, (short)0, acc[ni], false, false);            \
                }                                                                        \
            }                                                                            \
            __syncthreads();                                                             \
        }                                                                                \
    }

#if HAVE_TDM
#define STAGE_PROLOGUE(A_PTR, B_PTR, B_TENSOR_D1)                                        \
    const unsigned ldsA0 = (unsigned)(uintptr_t)&lA[0][0];                               \
    const unsigned ldsA1 = (unsigned)(uintptr_t)&lA[1][0];                               \
    const unsigned ldsB0 = (unsigned)(uintptr_t)&lB[0][0];                               \
    const unsigned ldsB1 = (unsigned)(uintptr_t)&lB[1][0];                               \
    if (wave == 0) {                                                                     \
        tdm_load_2d(ldsA0, (A_PTR), TM, TM, D_SZ);                                       \
        tdm_load_2d(ldsB0, (B_PTR), (B_TENSOR_D1), TN, D_SZ);                            \
    }
#define STAGE_WAIT(ch)                                                                   \
    if (wave == 0) __builtin_amdgcn_s_wait_tensorcnt(0)
#define STAGE_ISSUE_NEXT(ch, A_PTR, B_PTR, B_TENSOR_D1)                                  \
    if ((ch) + 1 < NCH && wave == 0) {                                                   \
        const int k1 = ((ch) + 1) * TK;                                                  \
        tdm_load_2d(((ch) & 1) ? ldsA0 : ldsA1, (A_PTR) + k1, TM, TM, D_SZ);             \
        tdm_load_2d(((ch) & 1) ? ldsB0 : ldsB1, (B_PTR) + k1, (B_TENSOR_D1), TN, D_SZ);  \
    }
#else
#define STAGE_PROLOGUE(A_PTR, B_PTR, B_TENSOR_D1)                                        \
    do { } while (0);
#define STAGE_WAIT(ch)                                                                   \
    do {                                                                                 \
        const int k0 = (ch) * TK;                                                        \
        const int cur0 = (ch) & 1;                                                       \
        __syncthreads();                                                                 \
        _Pragma("unroll")                                                                \
        for (int i = 0; i < 4; ++i) {                                                    \
            int lin = tid + i * 256;                                                     \
            int row = lin >> 3, c4 = lin & 7;                                            \
            float4 v = *(const float4*)(gemmA + (size_t)row * D_SZ + k0 + c4 * 4);       \
            *(float4*)(&lA[cur0][row * LDSS + c4 * 4]) = v;                              \
        }                                                                                \
        _Pragma("unroll")                                                                \
        for (int i = 0; i < 2; ++i) {                                                    \
            int lin = tid + i * 256;                                                     \
            int row = lin >> 3, c4 = lin & 7;                                            \
            float4 v = make_float4(0.f, 0.f, 0.f, 0.f);                                  \
            if ((unsigned)row < gemmBrows)                                               \
                v = *(const float4*)(gemmB + (size_t)row * D_SZ + k0 + c4 * 4);          \
            *(float4*)(&lB[cur0][row * LDSS + c4 * 4]) = v;                              \
        }                                                                                \
    } while (0)
#define STAGE_ISSUE_NEXT(ch, A_PTR, B_PTR, B_TENSOR_D1) do { } while (0)
#endif

// ---------------------------------------------------------------------------
// SYRK via fp32 WMMA: G = X @ X^T   (2048 x 2048)
// ---------------------------------------------------------------------------
__global__ __launch_bounds__(256) void wmma_syrk_kernel(const float* __restrict__ X,
                                                        float* __restrict__ G) {
    __shared__ float lA[2][TM * LDSS];
    __shared__ float lB[2][TN * LDSS];
    const int tid  = threadIdx.x;
    const int lane = tid & 31;
    const int wave = tid >> 5;
    const int nb = blockIdx.x * TN;
    const int mb = blockIdx.y * TM;

    const float* gemmA = X + (size_t)mb * D_SZ;
    const float* gemmB = X + (size_t)nb * D_SZ;
    const unsigned gemmBrows = TN;
    (void)gemmA; (void)gemmB; (void)gemmBrows;

    STAGE_PROLOGUE(gemmA, gemmB, TN)
    GEMM_TILE_LOOP(gemmA, gemmB, TN)

    // C/D layout: vgpr r, lanes 0-15 -> M=r ; lanes 16-31 -> M=r+8 ; N = lane&15
    const int rbase = mb + wave * 16 + 8 * (lane >> 4);
    const int col0  = nb + (lane & 15);
    #pragma unroll
    for (int ni = 0; ni < 4; ++ni)
        #pragma unroll
        for (int r = 0; r < 8; ++r)
            G[(size_t)(rbase + r) * B_SZ + col0 + ni * 16] = acc[ni][r];
}

// ---------------------------------------------------------------------------
// ArcFace GEMM + fused split-softmax partials.
// ---------------------------------------------------------------------------
__global__ __launch_bounds__(256) void wmma_arcface_kernel(
        const float* __restrict__ X, const float* __restrict__ W,
        const float* __restrict__ finv, const float* __restrict__ winv,
        const int* __restrict__ labels,
        float* __restrict__ m_part, float* __restrict__ s_part,
        float* __restrict__ tgt) {
    __shared__ float lA[2][TM * LDSS];
    __shared__ float lB[2][TN * LDSS];
    __shared__ float sFinv[TM];
    __shared__ int   sLab[TM];
    __shared__ float sWinv[TN];

    const int tid  = threadIdx.x;
    const int lane = tid & 31;
    const int wave = tid >> 5;
    const int tileN = blockIdx.x;
    const int cb = tileN * TN;
    const int mb = blockIdx.y * TM;

    if (tid < TM) { sFinv[tid] = finv[mb + tid]; sLab[tid] = labels[mb + tid]; }
    if (tid >= TM && tid < TM + TN) {
        int c = cb + (tid - TM);
        sWinv[tid - TM] = (c < C_SZ) ? winv[c] : 0.f;
    }

    const float* gemmA = X + (size_t)mb * D_SZ;
    const float* gemmB = W + (size_t)cb * D_SZ;
    const unsigned validRows = (unsigned)(C_SZ - cb) < (unsigned)TN ? (unsigned)(C_SZ - cb) : (unsigned)TN;
    const unsigned gemmBrows = validRows;
    (void)gemmA; (void)gemmB; (void)gemmBrows;

    STAGE_PROLOGUE(gemmA, gemmB, validRows)
    GEMM_TILE_LOOP(gemmA, gemmB, validRows)

    // ---- fused post-processing: logits, margin, per-row split-softmax ----
    const int half  = lane >> 4;
    const int nlane = lane & 15;
    const int rloc0 = wave * 16 + 8 * half;
    #pragma unroll
    for (int r = 0; r < 8; ++r) {
        const int lrow = rloc0 + r;
        const float fi = sFinv[lrow];
        const int   lb = sLab[lrow];
        float lg[4];
        #pragma unroll
        for (int ni = 0; ni < 4; ++ni) {
            int c = cb + ni * 16 + nlane;
            float cv = acc[ni][r] * fi * sWinv[ni * 16 + nlane];
            cv = fminf(fmaxf(cv, -1.f + 1e-7f), 1.f - 1e-7f);
            float sv = sqrtf(1.f - cv * cv + 1e-7f);
            float lt = (c == lb) ? (cv * COS_M_C - sv * SIN_M_C) : cv;
            float logit = SCALE_S * lt;
            if (c >= C_SZ) logit = -1e30f;
            if (c == lb) tgt[mb + lrow] = logit;   // exactly one writer grid-wide
            lg[ni] = logit;
        }
        float m = fmaxf(fmaxf(lg[0], lg[1]), fmaxf(lg[2], lg[3]));
        float s = __expf(lg[0] - m) + __expf(lg[1] - m) +
                  __expf(lg[2] - m) + __expf(lg[3] - m);
        #pragma unroll
        for (int off = 1; off < 16; off <<= 1) {   // stays within the 16-lane half
            float m2 = __shfl_xor(m, off, 32);
            float s2 = __shfl_xor(s, off, 32);
            float M  = fmaxf(m, m2);
            s = s * __expf(m - M) + s2 * __expf(m2 - M);
            m = M;
        }
        if (nlane == 0) {
            m_part[(size_t)(mb + lrow) * NT_ARC + tileN] = m;
            s_part[(size_t)(mb + lrow) * NT_ARC + tileN] = s;
        }
    }
}

// ---------------------------------------------------------------------------
// Batch-hard triplet per row, from Gram matrix
// ---------------------------------------------------------------------------
__global__ __launch_bounds__(256) void triplet_kernel(const float* __restrict__ G,
                                                      const float* __restrict__ sq,
                                                      const int* __restrict__ labels,
                                                      float* __restrict__ trip) {
    __shared__ float rp[256];
    __shared__ float rn[256];
    const int i = blockIdx.x;
    const int tid = threadIdx.x;
    const float sqi = sq[i];
    const int li = labels[i];
    float hp = 0.f, hn = 1e30f;
    for (int j = tid; j < B_SZ; j += 256) {
        float d2 = sqi + sq[j] - 2.f * G[(size_t)i * B_SZ + j];
        float d  = d2 > 0.f ? sqrtf(d2) : 0.f;
        bool pos = (labels[j] == li);
        hp = fmaxf(hp, pos ? d : 0.f);
        hn = fminf(hn, d + (pos ? 1e6f : 0.f));
    }
    rp[tid] = hp; rn[tid] = hn;
    __syncthreads();
    for (int off = 128; off > 0; off >>= 1) {
        if (tid < off) {
            rp[tid] = fmaxf(rp[tid], rp[tid + off]);
            rn[tid] = fminf(rn[tid], rn[tid + off]);
        }
        __syncthreads();
    }
    if (tid == 0) trip[i] = fmaxf(rp[0] - rn[0] + TRIP_MARGIN, 0.f);
}

// ---------------------------------------------------------------------------
// Center loss per row
// ---------------------------------------------------------------------------
__global__ __launch_bounds__(256) void center_kernel(const float* __restrict__ X,
                                                     const float* __restrict__ centers,
                                                     const int* __restrict__ labels,
                                                     float* __restrict__ cent) {
    __shared__ float red[256];
    const int i = blockIdx.x;
    const int tid = threadIdx.x;
    const int lb = labels[i];
    float s = 0.f;
    for (int d = tid; d < D_SZ; d += 256) {
        float v = X[(size_t)i * D_SZ + d] - centers[(size_t)lb * D_SZ + d];
        s += v * v;
    }
    red[tid] = s;
    __syncthreads();
    for (int off = 128; off > 0; off >>= 1) {
        if (tid < off) red[tid] += red[tid + off];
        __syncthreads();
    }
    if (tid == 0) cent[i] = red[0];
}

// ---------------------------------------------------------------------------
// Combine split-softmax partials per row -> -log p[target]
// ---------------------------------------------------------------------------
__global__ __launch_bounds__(256) void arcfin_kernel(const float* __restrict__ m_part,
                                                     const float* __restrict__ s_part,
                                                     const float* __restrict__ tgt,
                                                     float* __restrict__ arc) {
    __shared__ float sm[256];
    __shared__ float ss[256];
    const int i = blockIdx.x;
    const int tid = threadIdx.x;
    float m = -1e30f, s = 0.f;
    for (int t = tid; t < NT_ARC; t += 256) {
        float m2 = m_part[(size_t)i * NT_ARC + t];
        float s2 = s_part[(size_t)i * NT_ARC + t];
        float M = fmaxf(m, m2);
        s = s * __expf(m - M) + s2 * __expf(m2 - M);
        m = M;
    }
    sm[tid] = m; ss[tid] = s;
    __syncthreads();
    for (int off = 128; off > 0; off >>= 1) {
        if (tid < off) {
            float m2 = sm[tid + off], s2 = ss[tid + off];
            float M = fmaxf(sm[tid], m2);
            ss[tid] = ss[tid] * __expf(sm[tid] - M) + s2 * __expf(m2 - M);
            sm[tid] = M;
        }
        __syncthreads();
    }
    if (tid == 0) arc[i] = -(tgt[i] - (sm[0] + logf(ss[0])));
}

// ---------------------------------------------------------------------------
// Final deterministic reduce
// ---------------------------------------------------------------------------
__global__ __launch_bounds__(256) void final_kernel(const float* __restrict__ trip,
                                                    const float* __restrict__ cent,
                                                    const float* __restrict__ arc,
                                                    float* __restrict__ out) {
    __shared__ float r0[256], r1[256], r2[256];
    const int tid = threadIdx.x;
    float a = 0.f, b = 0.f, c = 0.f;
    for (int i = tid; i < B_SZ; i += 256) {
        a += trip[i]; b += cent[i]; c += arc[i];
    }
    r0[tid] = a; r1[tid] = b; r2[tid] = c;
    __syncthreads();
    for (int off = 128; off > 0; off >>= 1) {
        if (tid < off) {
            r0[tid] += r0[tid + off];
            r1[tid] += r1[tid + off];
            r2[tid] += r2[tid + off];
        }
        __syncthreads();
    }
    if (tid == 0) {
        const float invB = 1.0f / (float)B_SZ;
        out[0] = 1.0f * (r0[0] * invB) + 0.1f * (r1[0] * invB) + 0.1f * (r2[0] * invB);
    }
}

extern "C" void kernel_launch(void* const* d_in, const int* in_sizes, int n_in,
                              void* d_out, int out_size, void* d_ws, size_t ws_size,
                              hipStream_t stream) {
    const float* features = (const float*)d_in[0];
    const int*   labels   = (const int*)d_in[1];
    const float* arc_w    = (const float*)d_in[2];
    const float* centers  = (const float*)d_in[3];
    float* out = (float*)d_out;

    // workspace layout (floats), ~30 MB total
    float* ws     = (float*)d_ws;
    float* sq     = ws;                                   // B
    float* finv   = sq + B_SZ;                            // B
    float* winv   = finv + B_SZ;                          // C
    float* G      = winv + C_SZ;                          // B*B
    float* m_part = G + (size_t)B_SZ * B_SZ;              // B*NT
    float* s_part = m_part + (size_t)B_SZ * NT_ARC;       // B*NT
    float* tgt    = s_part + (size_t)B_SZ * NT_ARC;       // B
    float* trip   = tgt + B_SZ;                           // B
    float* cent   = trip + B_SZ;                          // B
    float* arc    = cent + B_SZ;                          // B

    rownorm_kernel<<<B_SZ, 256, 0, stream>>>(features, finv, sq);
    rownorm_kernel<<<C_SZ, 256, 0, stream>>>(arc_w, winv, nullptr);

    dim3 gs(B_SZ / TN, B_SZ / TM);
    wmma_syrk_kernel<<<gs, 256, 0, stream>>>(features, G);

    dim3 ga(NT_ARC, B_SZ / TM);
    wmma_arcface_kernel<<<ga, 256, 0, stream>>>(features, arc_w, finv, winv, labels,
                                                m_part, s_part, tgt);

    triplet_kernel<<<B_SZ, 256, 0, stream>>>(G, sq, labels, trip);
    center_kernel<<<B_SZ, 256, 0, stream>>>(features, centers, labels, cent);
    arcfin_kernel<<<B_SZ, 256, 0, stream>>>(m_part, s_part, tgt, arc);
    final_kernel<<<1, 256, 0, stream>>>(trip, cent, arc, out);
}